// NTMMemory_75359496175674
// MI455X (gfx1250) — compile-verified
//
#include <hip/hip_runtime.h>
#include <math.h>

#define NROWS 2000000
#define NCOLS 128

typedef __attribute__((ext_vector_type(2))) float v2f;
typedef __attribute__((ext_vector_type(8))) float v8f;

__device__ __forceinline__ float block_reduce_max(float v, float* red) {
  int tid = threadIdx.x;
  red[tid] = v;
  __syncthreads();
  for (int s = 128; s > 0; s >>= 1) {
    if (tid < s) red[tid] = fmaxf(red[tid], red[tid + s]);
    __syncthreads();
  }
  return red[0];
}

__device__ __forceinline__ float block_reduce_sum(float v, float* red) {
  int tid = threadIdx.x;
  red[tid] = v;
  __syncthreads();
  for (int s = 128; s > 0; s >>= 1) {
    if (tid < s) red[tid] += red[tid + s];
    __syncthreads();
  }
  return red[0];
}

// ---------------------------------------------------------------------------
// Kernel 1: a[i] = beta * (1 + cos(mem_i, k)), block-partial max.
// Each wave owns a 16-row tile; the 128-deep dot product is computed with
// 32 chained V_WMMA_F32_16X16X4_F32 ops (B columns all equal to the k-chunk,
// so D[m][*] = dot(row m, k)). Row sqnorm runs on VALU alongside.
// A-frag layout (ISA 7.12.2): lanes0-15 carry K=k0,k0+1; lanes16-31 K=k0+2,k0+3.
// D layout: lane0 VGPR v -> row v; lane16 VGPR v -> row v+8.
// ---------------------------------------------------------------------------
__global__ __launch_bounds__(256) void k_sim(const float* __restrict__ mem,
                                             const float* __restrict__ kvec,
                                             const float* __restrict__ beta_p,
                                             float* __restrict__ arr,
                                             float* __restrict__ pmax) {
  __shared__ float kq[NCOLS];
  __shared__ float dot_lds[128];  // 8 waves * 16 rows
  __shared__ float red[256];
  const int tid = threadIdx.x;
  const int lane = tid & 31;
  const int wave = tid >> 5;
  if (tid < NCOLS) kq[tid] = kvec[tid] + 1e-16f;
  __syncthreads();

  float kn2 = 0.f;
  for (int i = 0; i < NCOLS; ++i) { float t = kq[i]; kn2 += t * t; }
  const float kn = fmaxf(sqrtf(kn2), 1e-12f);
  const float beta = beta_p[0];

  const int row0 = blockIdx.x * 128 + wave * 16;
  const int m = lane & 15;
  const int hi2 = (lane >= 16) ? 2 : 0;
  const float* rowp = mem + (size_t)(row0 + m) * NCOLS + hi2;
  __builtin_prefetch(rowp, 0, 1);  // global_prefetch_b8 (streaming 1 GB)

  v8f acc = {};
  float sq = 0.f;
#pragma unroll
  for (int k0 = 0; k0 < NCOLS; k0 += 4) {
    v2f av = *(const v2f*)(rowp + k0);
    av.x += 1e-16f;
    av.y += 1e-16f;
    v2f bv;
    bv.x = kq[k0 + hi2];
    bv.y = kq[k0 + hi2 + 1];
    sq += av.x * av.x + av.y * av.y;
    acc = __builtin_amdgcn_wmma_f32_16x16x4_f32(false, av, false, bv, (short)0,
                                                acc, false, false);
  }
  // lane L holds k%4 in {0,1} partial of row L; lane L+16 the {2,3} partial
  sq += __shfl_xor(sq, 16, 32);

  if (lane == 0) {
#pragma unroll
    for (int v = 0; v < 8; ++v) dot_lds[wave * 16 + v] = acc[v];
  } else if (lane == 16) {
#pragma unroll
    for (int v = 0; v < 8; ++v) dot_lds[wave * 16 + 8 + v] = acc[v];
  }
  __syncthreads();

  float aval = -3.0e38f;
  if (lane < 16) {
    float dot = dot_lds[wave * 16 + lane];
    float rn = fmaxf(sqrtf(sq), 1e-12f);
    aval = beta * (1.f + dot / (rn * kn));
    arr[row0 + lane] = aval;
  }
  float bm = block_reduce_max(aval, red);
  if (tid == 0) pmax[blockIdx.x] = bm;
}

// mode 0: max; mode 1: sum; mode 2: 1/(sum + 1e-16)
__global__ __launch_bounds__(256) void k_reduce(const float* __restrict__ p,
                                                int n, float* __restrict__ outp,
                                                int mode) {
  __shared__ float red[256];
  float v = (mode == 0) ? -3.0e38f : 0.f;
  for (int i = threadIdx.x; i < n; i += 256) {
    float x = p[i];
    v = (mode == 0) ? fmaxf(v, x) : (v + x);
  }
  float r = (mode == 0) ? block_reduce_max(v, red) : block_reduce_sum(v, red);
  if (threadIdx.x == 0) outp[0] = (mode == 2) ? (1.f / (r + 1e-16f)) : r;
}

__global__ __launch_bounds__(256) void k_exp(float* __restrict__ arr,
                                             const float* __restrict__ scal,
                                             float* __restrict__ psum) {
  __shared__ float red[256];
  const float gmax = scal[0];
  float s = 0.f;
  for (int i = blockIdx.x * 256 + threadIdx.x; i < NROWS;
       i += 256 * gridDim.x) {
    float e = expf(arr[i] - gmax);
    arr[i] = e;
    s += e;
  }
  float bs = block_reduce_sum(s, red);
  if (threadIdx.x == 0) psum[blockIdx.x] = bs;
}

// wg(j) = g*e[j]/Z + (1-g)*w_prev[j]; w_hat = s0*wg(i-1)+s1*wg(i)+s2*wg(i+1)
// (circular); out = w_hat^gamma; block-partial sum.
__global__ __launch_bounds__(256) void k_shift(
    const float* __restrict__ e, const float* __restrict__ w_prev,
    const float* __restrict__ g_p, const float* __restrict__ s_p,
    const float* __restrict__ gamma_p, const float* __restrict__ scal,
    float* __restrict__ outw, float* __restrict__ psum) {
  __shared__ float red[256];
  const float g = g_p[0];
  const float s0 = s_p[0], s1 = s_p[1], s2 = s_p[2];
  const float gamma = gamma_p[0];
  const float invZ = 1.f / scal[1];
  const float omg = 1.f - g;
  float acc = 0.f;
  for (int i = blockIdx.x * 256 + threadIdx.x; i < NROWS;
       i += 256 * gridDim.x) {
    int ip = (i == 0) ? (NROWS - 1) : (i - 1);
    int in2 = (i == NROWS - 1) ? 0 : (i + 1);
    float wgp = g * e[ip] * invZ + omg * w_prev[ip];
    float wgc = g * e[i] * invZ + omg * w_prev[i];
    float wgn = g * e[in2] * invZ + omg * w_prev[in2];
    float wh = s0 * wgp + s1 * wgc + s2 * wgn;
    float w = powf(wh, gamma);
    outw[i] = w;
    acc += w;
  }
  float bs = block_reduce_sum(acc, red);
  if (threadIdx.x == 0) psum[blockIdx.x] = bs;
}

__global__ __launch_bounds__(256) void k_scale(float* __restrict__ outw,
                                               const float* __restrict__ invS) {
  const float s = invS[0];
  for (int i = blockIdx.x * 256 + threadIdx.x; i < NROWS;
       i += 256 * gridDim.x)
    outw[i] *= s;
}

extern "C" void kernel_launch(void* const* d_in, const int* in_sizes, int n_in,
                              void* d_out, int out_size, void* d_ws,
                              size_t ws_size, hipStream_t stream) {
  (void)in_sizes; (void)n_in; (void)out_size; (void)ws_size;
  const float* mem    = (const float*)d_in[0];
  const float* k      = (const float*)d_in[1];
  const float* beta   = (const float*)d_in[2];
  const float* g      = (const float*)d_in[3];
  const float* s      = (const float*)d_in[4];
  const float* gamma  = (const float*)d_in[5];
  const float* w_prev = (const float*)d_in[6];
  float* out = (float*)d_out;

  // Workspace layout: arr[2M] (a -> e in place), pmax[15632], psum[2048],
  // psum2[2048], scal[4]  ~= 8.08 MB
  float* arr = (float*)d_ws;
  const int NB1 = NROWS / 128;  // 15625, exact
  float* pmax = arr + NROWS;
  float* psum = pmax + ((NB1 + 15) & ~15);
  const int NB2 = 2048;
  float* psum2 = psum + NB2;
  float* scal = psum2 + NB2;  // [0]=gmax [1]=Z [2]=1/(S+1e-16)

  k_sim<<<NB1, 256, 0, stream>>>(mem, k, beta, arr, pmax);
  k_reduce<<<1, 256, 0, stream>>>(pmax, NB1, scal + 0, 0);
  k_exp<<<NB2, 256, 0, stream>>>(arr, scal, psum);
  k_reduce<<<1, 256, 0, stream>>>(psum, NB2, scal + 1, 1);
  k_shift<<<NB2, 256, 0, stream>>>(arr, w_prev, g, s, gamma, scal, out, psum2);
  k_reduce<<<1, 256, 0, stream>>>(psum2, NB2, scal + 2, 2);
  k_scale<<<NB2, 256, 0, stream>>>(out, scal + 2);
}